// GAT_37194416783905
// MI455X (gfx1250) — compile-verified
//
#include <hip/hip_runtime.h>
#include <hip/hip_bf16.h>
#include <math.h>

// ---------------------------------------------------------------------------
// GAT (2-layer, PyG semantics) for gfx1250 / MI455X.
// Dense GEMMs: V_WMMA_F32_16X16X4_F32 (exact f32), A-tile + full B staged in
// LDS (320KB/WGP), branch-free fully-unrolled 32-WMMA inner loop fed by
// ds_load_b64. Sparse phase: 3-pass segment softmax, u32-encoded float atomic
// max + f32 atomic adds (working set is L2-resident on the 192MB L2).
// Workspace requirement: (2*N*128 + 4*N*8) * 4 bytes ~= 35 MB.
// ---------------------------------------------------------------------------

typedef __attribute__((ext_vector_type(2))) float v2f;
typedef __attribute__((ext_vector_type(8))) float v8f;

#define IN_DIM 128
#define H_DIM 128
#define HEADS1 8
#define DH1 16
#define OUT_DIM 64
#define NEG_SLOPE 0.2f
#define EPS_DENOM 1e-16f

// ---------------- float <-> monotonic u32 for atomic max --------------------
__device__ inline unsigned enc_f32(float f) {
  unsigned u = __float_as_uint(f);
  return u ^ (unsigned)(((int)u >> 31) | 0x80000000);
}
__device__ inline float dec_u32(unsigned k) {
  unsigned u = (k & 0x80000000u) ? (k ^ 0x80000000u) : ~k;
  return __uint_as_float(u);
}

// ---------------- WMMA GEMM: C[M x NCOLS] = A[M x 128] * B[128 x NCOLS] -----
// Block = NCOLS/16 waves; wave w owns the 16x16 output tile at columns
// [16w, 16w+16). A tile (16x128) and all of B staged in LDS once per block.
// B is stored pair-interleaved: sB pair p=(k>>1), col c holds {B[2p][c],
// B[2p+1][c]} so each lane's B fragment is one 64-bit LDS read.
template <int NCOLS>
__global__ void __launch_bounds__((NCOLS / 16) * 32)
wmma_gemm_k128(const float* __restrict__ A, const float* __restrict__ B,
               float* __restrict__ C, int M) {
  __shared__ float sA[16 * 128];          // 8 KB
  __shared__ v2f   sB[64 * NCOLS];        // 128*NCOLS floats (64/32 KB)

  const int tid   = threadIdx.x;
  const int nthr  = (NCOLS / 16) * 32;
  const int row0  = blockIdx.x * 16;

  // ---- fill A tile (float4 granularity, guard only here) ----
  for (int idx = tid; idx < 16 * 32; idx += nthr) {   // 16 rows * 32 float4
    const int r  = idx >> 5;
    const int c4 = idx & 31;
    const int gr = row0 + r;
    float4 v = (gr < M) ? ((const float4*)A)[(size_t)gr * 32 + c4]
                        : make_float4(0.f, 0.f, 0.f, 0.f);
    ((float4*)sA)[idx] = v;
  }
  // ---- fill B (pair-interleaved) ----
  for (int idx = tid; idx < 64 * NCOLS; idx += nthr) {
    const int kp = idx / NCOLS;           // K pair 0..63 (shift: NCOLS pow2)
    const int c  = idx % NCOLS;
    v2f b;
    b.x = B[(size_t)(2 * kp)     * NCOLS + c];
    b.y = B[(size_t)(2 * kp + 1) * NCOLS + c];
    sB[idx] = b;
  }
  __syncthreads();

  const int lane = tid & 31;
  const int wave = tid >> 5;
  const int half = lane >> 4;             // 0: lanes 0-15, 1: lanes 16-31
  const int l15  = lane & 15;
  const int c    = wave * 16 + l15;       // this lane's output/B column

  v8f acc = {};
#pragma unroll
  for (int k0 = 0; k0 < 128; k0 += 4) {
    const int k = k0 + half * 2;          // lane's first K index (even)
    v2f a = *(const v2f*)&sA[l15 * 128 + k];    // A[l15][k..k+1]
    v2f b = sB[(k >> 1) * NCOLS + c];           // B[k][c], B[k+1][c]
    acc = __builtin_amdgcn_wmma_f32_16x16x4_f32(
        /*neg_a=*/false, a, /*neg_b=*/false, b,
        /*c_mod=*/(short)0, acc, /*reuse_a=*/false, /*reuse_b=*/false);
  }
#pragma unroll
  for (int i = 0; i < 8; ++i) {
    const int r = row0 + i + half * 8;
    if (r < M) C[(size_t)r * NCOLS + c] = acc[i];
  }
}

// ---------------- per-node attention logits --------------------------------
__global__ void node_alpha(const float* __restrict__ Hm,
                           const float* __restrict__ a_src,
                           const float* __restrict__ a_dst,
                           float* __restrict__ as, float* __restrict__ ad,
                           int N, int heads, int D) {
  const int idx = blockIdx.x * blockDim.x + threadIdx.x;
  if (idx >= N * heads) return;
  const int n = idx / heads, h = idx % heads;
  const float* row = Hm + (size_t)n * heads * D + (size_t)h * D;
  float s = 0.0f, d = 0.0f;
  for (int i = 0; i < D; ++i) {
    s += row[i] * a_src[h * D + i];
    d += row[i] * a_dst[h * D + i];
  }
  as[idx] = s;
  ad[idx] = d;
}

// ---------------- buffer init ----------------------------------------------
__global__ void init_buffers(unsigned* __restrict__ emax,
                             float* __restrict__ denom,
                             float* __restrict__ out, int NH, int NF) {
  const int i = blockIdx.x * blockDim.x + threadIdx.x;
  if (i < NH) { emax[i] = 0u; denom[i] = 0.0f; }
  if (i < NF) { out[i] = 0.0f; }
}

// ---------------- edge helpers ---------------------------------------------
__device__ inline void edge_sd(const int* __restrict__ ei, int e, int E,
                               int& s, int& d) {
  if (e < E) { s = ei[e]; d = ei[E + e]; }
  else       { s = e - E; d = s; }       // synthesized self-loop
}
__device__ inline float leaky(float v) {
  return v > 0.0f ? v : NEG_SLOPE * v;
}

// ---------------- pass 1: segment max --------------------------------------
__global__ void edge_max_k(const int* __restrict__ ei, int E, int Etot,
                           const float* __restrict__ as,
                           const float* __restrict__ ad,
                           unsigned* __restrict__ emax, int heads) {
  const int t = blockIdx.x * blockDim.x + threadIdx.x;
  if (t >= Etot * heads) return;
  const int e = t / heads, h = t % heads;
  int s, d; edge_sd(ei, e, E, s, d);
  const float v = leaky(as[s * heads + h] + ad[d * heads + h]);
  atomicMax(&emax[d * heads + h], enc_f32(v));
}

// ---------------- pass 2: exp-sum ------------------------------------------
__global__ void edge_expsum_k(const int* __restrict__ ei, int E, int Etot,
                              const float* __restrict__ as,
                              const float* __restrict__ ad,
                              const unsigned* __restrict__ emax,
                              float* __restrict__ denom, int heads) {
  const int t = blockIdx.x * blockDim.x + threadIdx.x;
  if (t >= Etot * heads) return;
  const int e = t / heads, h = t % heads;
  int s, d; edge_sd(ei, e, E, s, d);
  const float v = leaky(as[s * heads + h] + ad[d * heads + h]);
  const float m = dec_u32(emax[d * heads + h]);
  atomicAdd(&denom[d * heads + h], expf(v - m));
}

// ---------------- pass 3: weighted scatter ---------------------------------
__global__ void edge_scatter_k(const int* __restrict__ ei, int E, int Etot,
                               const float* __restrict__ as,
                               const float* __restrict__ ad,
                               const unsigned* __restrict__ emax,
                               const float* __restrict__ denom,
                               const float* __restrict__ Hm,
                               float* __restrict__ out, int heads, int D) {
  const int t = blockIdx.x * blockDim.x + threadIdx.x;
  if (t >= Etot * heads) return;
  const int e = t / heads, h = t % heads;
  int s, d; edge_sd(ei, e, E, s, d);
  const float v = leaky(as[s * heads + h] + ad[d * heads + h]);
  const float m = dec_u32(emax[d * heads + h]);
  const float alpha = expf(v - m) / (denom[d * heads + h] + EPS_DENOM);
  const float* hrow = Hm + (size_t)s * heads * D + (size_t)h * D;
  float* orow = out + (size_t)d * heads * D + (size_t)h * D;
  for (int i = 0; i < D; ++i) atomicAdd(&orow[i], hrow[i] * alpha);
}

// ---------------- bias + ELU (elementwise, in-place safe), F=128 -----------
__global__ void bias_elu_k(const float* __restrict__ in,
                           const float* __restrict__ b,
                           float* __restrict__ out, int total) {
  const int t = blockIdx.x * blockDim.x + threadIdx.x;
  if (t >= total) return;
  const float v = in[t] + b[t & (H_DIM - 1)];
  out[t] = v > 0.0f ? v : (expf(v) - 1.0f);
}

// ---------------- bias + ELU + log_softmax over 64 features ----------------
__global__ void bias_elu_logsoftmax_k(const float* __restrict__ in,
                                      const float* __restrict__ b,
                                      float* __restrict__ out, int N) {
  const int n = blockIdx.x * blockDim.x + threadIdx.x;
  if (n >= N) return;
  constexpr int F = OUT_DIM;
  float vals[F];
  float mx = -INFINITY;
#pragma unroll
  for (int f = 0; f < F; ++f) {
    float v = in[(size_t)n * F + f] + b[f];
    v = v > 0.0f ? v : (expf(v) - 1.0f);
    vals[f] = v;
    mx = fmaxf(mx, v);
  }
  float sum = 0.0f;
#pragma unroll
  for (int f = 0; f < F; ++f) sum += expf(vals[f] - mx);
  const float lse = mx + logf(sum);
#pragma unroll
  for (int f = 0; f < F; ++f) out[(size_t)n * F + f] = vals[f] - lse;
}

// ---------------------------------------------------------------------------
extern "C" void kernel_launch(void* const* d_in, const int* in_sizes, int n_in,
                              void* d_out, int out_size, void* d_ws, size_t ws_size,
                              hipStream_t stream) {
  const float* x      = (const float*)d_in[0];
  const int*   ei     = (const int*)d_in[1];     // [2, E] flat: src then dst
  const float* W1     = (const float*)d_in[2];   // [128, 128]
  const float* a_src1 = (const float*)d_in[3];   // [8, 16]
  const float* a_dst1 = (const float*)d_in[4];
  const float* b1     = (const float*)d_in[5];   // [128]
  const float* W2     = (const float*)d_in[6];   // [128, 64]
  const float* a_src2 = (const float*)d_in[7];   // [1, 64]
  const float* a_dst2 = (const float*)d_in[8];
  const float* b2     = (const float*)d_in[9];   // [64]
  float* out = (float*)d_out;

  const int N    = in_sizes[0] / IN_DIM;         // 30000
  const int E    = in_sizes[1] / 2;              // 480000
  const int Etot = E + N;                        // edges + self loops

  // ---- workspace layout (floats; emax buffers alias as u32) ----
  float* ws = (float*)d_ws;
  const size_t NF1 = (size_t)N * H_DIM;          // N*128
  float*    H1     = ws;                         // [N,128]
  float*    out1   = H1 + NF1;                   // [N,128], becomes x2 in-place
  float*    as1    = out1 + NF1;                 // [N,8]
  float*    ad1    = as1 + (size_t)N * HEADS1;   // [N,8]
  unsigned* emax1  = (unsigned*)(ad1 + (size_t)N * HEADS1);  // [N,8]
  float*    denom1 = (float*)(emax1 + (size_t)N * HEADS1);   // [N,8]
  // layer 2 reuses layer-1 regions (stream-ordered, layer 1 fully consumed)
  float*    H2     = H1;                         // [N,64]
  float*    out2   = H1 + (size_t)N * OUT_DIM;   // [N,64]
  float*    as2    = as1;                        // [N]
  float*    ad2    = ad1;
  unsigned* emax2  = emax1;
  float*    denom2 = denom1;

  const int rowTiles = (N + 15) / 16;            // 1875 (exact)
  const int TB = 256;
  auto cdiv = [](int a, int b) { return (a + b - 1) / b; };

  // ================= layer 1: heads=8, D=16, concat ==================
  wmma_gemm_k128<H_DIM><<<rowTiles, 256, 0, stream>>>(x, W1, H1, N);
  node_alpha<<<cdiv(N * HEADS1, TB), TB, 0, stream>>>(
      H1, a_src1, a_dst1, as1, ad1, N, HEADS1, DH1);
  init_buffers<<<cdiv((int)NF1, TB), TB, 0, stream>>>(
      emax1, denom1, out1, N * HEADS1, (int)NF1);
  edge_max_k<<<cdiv(Etot * HEADS1, TB), TB, 0, stream>>>(
      ei, E, Etot, as1, ad1, emax1, HEADS1);
  edge_expsum_k<<<cdiv(Etot * HEADS1, TB), TB, 0, stream>>>(
      ei, E, Etot, as1, ad1, emax1, denom1, HEADS1);
  edge_scatter_k<<<cdiv(Etot * HEADS1, TB), TB, 0, stream>>>(
      ei, E, Etot, as1, ad1, emax1, denom1, H1, out1, HEADS1, DH1);
  bias_elu_k<<<cdiv((int)NF1, TB), TB, 0, stream>>>(
      out1, b1, out1, (int)NF1);                 // x2 = elu(out1+b1), in-place

  // ================= layer 2: heads=1, D=64, mean (=identity) ========
  wmma_gemm_k128<OUT_DIM><<<rowTiles, 128, 0, stream>>>(out1, W2, H2, N);
  node_alpha<<<cdiv(N, TB), TB, 0, stream>>>(
      H2, a_src2, a_dst2, as2, ad2, N, 1, OUT_DIM);
  init_buffers<<<cdiv(N * OUT_DIM, TB), TB, 0, stream>>>(
      emax2, denom2, out2, N, N * OUT_DIM);
  edge_max_k<<<cdiv(Etot, TB), TB, 0, stream>>>(
      ei, E, Etot, as2, ad2, emax2, 1);
  edge_expsum_k<<<cdiv(Etot, TB), TB, 0, stream>>>(
      ei, E, Etot, as2, ad2, emax2, denom2, 1);
  edge_scatter_k<<<cdiv(Etot, TB), TB, 0, stream>>>(
      ei, E, Etot, as2, ad2, emax2, denom2, H2, out2, 1, OUT_DIM);
  bias_elu_logsoftmax_k<<<cdiv(N, 128), 128, 0, stream>>>(out2, b2, out, N);
}